// AttentionSelector_5703716569098
// MI455X (gfx1250) — compile-verified
//
#include <hip/hip_runtime.h>
#include <hip/hip_bf16.h>

typedef __attribute__((ext_vector_type(2))) float v2f;
typedef __attribute__((ext_vector_type(8))) float v8f;

namespace {
constexpr int D     = 690;        // feature dim
constexpr int DP    = 692;        // D padded to multiple of 4 for the WMMA K-loop
constexpr int R     = 53;         // relations (output cols), padded to 64 across 4 waves
constexpr int NPAIR = 345;        // D / 2 (rows are 8B aligned -> b64 loads are safe)
constexpr int PAIRS = 11;         // ceil(345 / 32) pair-slots per lane
constexpr int BAGS_PER_BLOCK = 16;
constexpr int THREADS = 128;      // 4 wave32 waves
}

__global__ __launch_bounds__(THREADS)
void bag_attention_fused(const float* __restrict__ repre,
                         const float* __restrict__ relmat,
                         const float* __restrict__ bias,
                         const int*   __restrict__ scope,
                         const int*   __restrict__ labels,
                         float* __restrict__ out,
                         int num_bags)
{
    __shared__ float att[BAGS_PER_BLOCK][DP];

    const int lane    = threadIdx.x & 31;
    const int wave    = threadIdx.x >> 5;
    const int bagBase = blockIdx.x * BAGS_PER_BLOCK;

    // ---------------- Phase 1: streaming online-softmax attention ----------
    // Each wave owns 4 bags; repre rows are read exactly once (flash-style
    // rescaling). Loads are paired (global_load_b64): 11 per matrix per row.
    for (int j = 0; j < 4; ++j) {
        const int row = wave * 4 + j;
        const int bag = bagBase + row;

        v2f acc[PAIRS];
#pragma unroll
        for (int c = 0; c < PAIRS; ++c) acc[c] = (v2f)0.0f;
        float m = -__builtin_inff();
        float s = 0.0f;

        if (bag < num_bags) {
            const int start = __builtin_amdgcn_readfirstlane(scope[bag * 2 + 0]);
            const int end   = __builtin_amdgcn_readfirstlane(scope[bag * 2 + 1]);
            for (int i = start; i < end; ++i) {
                const int lab = __builtin_amdgcn_readfirstlane(labels[i]);
                const v2f* rrow2 = (const v2f*)(repre  + (long)i   * D);
                const v2f* vrow2 = (const v2f*)(relmat + (long)lab * D);
                if (i + 1 < end) {
                    // one 128B line per lane covers the whole next row
                    __builtin_prefetch(repre + (long)(i + 1) * D + lane * 32, 0, 0);
                }
                v2f r[PAIRS];
                float partial = 0.0f;
#pragma unroll
                for (int c = 0; c < PAIRS; ++c) {
                    const int p  = lane + 32 * c;
                    const int pc = (p < NPAIR) ? p : (NPAIR - 1);   // clamp addr
                    v2f rr = __builtin_nontemporal_load(rrow2 + pc); // streamed once
                    v2f vv = vrow2[pc];                              // L2-resident
                    if (p >= NPAIR) { rr = (v2f)0.0f; vv = (v2f)0.0f; } // mask pad
                    r[c] = rr;
                    partial = fmaf(rr.x, vv.x, fmaf(rr.y, vv.y, partial));
                }
                // wave32 butterfly reduction -> every lane holds the logit
#pragma unroll
                for (int off = 16; off > 0; off >>= 1)
                    partial += __shfl_xor(partial, off, 32);

                const float logit = partial;
                const float mNew  = fmaxf(m, logit);
                const float scale = expf(m - mNew);      // first iter: exp(-inf)=0
                const float w     = expf(logit - mNew);
                s = s * scale + w;
#pragma unroll
                for (int c = 0; c < PAIRS; ++c)
                    acc[c] = acc[c] * scale + r[c] * w;
                m = mNew;
            }
        }

        const float inv = (s > 0.0f) ? (1.0f / s) : 0.0f;
#pragma unroll
        for (int c = 0; c < PAIRS; ++c) {
            const int p = lane + 32 * c;
            if (p < NPAIR)
                *(v2f*)&att[row][2 * p] = acc[c] * inv;   // ds_store_b64, 8B aligned
        }
        if (lane < (DP - D)) att[row][D + lane] = 0.0f;   // zero K-pad columns
    }

    __syncthreads();

    // ---------------- Phase 2: 16x64 tile GEMM via V_WMMA_F32_16X16X4_F32 --
    // C(16 bags x 16 cols) = A(16 x K) * B(K x 16); wave w covers columns
    // [16w, 16w+16). A layout per ISA: lanes 0-15 -> {K,K+1}, lanes 16-31 ->
    // {K+2,K+3}; B is row-striped across lanes (mirror of the C layout).
    v8f c = {};
    const int colBase = wave * 16;
    const int mrow    = lane & 15;              // A row index / B column index
    const int ksel    = (lane < 16) ? 0 : 2;    // half-wave K pair select

    const int   bcol   = colBase + mrow;        // relation_mat row (output col)
    const bool  bvalid = (bcol < R);
    const float* bptr  = relmat + (long)((bcol < R) ? bcol : (R - 1)) * D;

#pragma unroll 4
    for (int kk = 0; kk < DP; kk += 4) {
        const int k0 = kk + ksel;               // even; 0 .. 690
        const int kc = (k0 < D - 1) ? k0 : (D - 2);   // clamp addr (even)
        v2f a = *(const v2f*)&att[mrow][k0];    // ds_load_b64 (pad cols are zero)
        v2f bv = *(const v2f*)(bptr + kc);      // global_load_b64, always in-bounds
        v2f b;
        b.x = (bvalid && (k0     < D)) ? bv.x : 0.0f;   // v_cndmask, no branches
        b.y = (bvalid && (k0 + 1 < D)) ? bv.y : 0.0f;
        // 8 args: (neg_a, A, neg_b, B, c_mod, C, reuse_a, reuse_b)
        c = __builtin_amdgcn_wmma_f32_16x16x4_f32(false, a, false, b,
                                                  (short)0, c, false, false);
    }

    // C layout: VGPR v, lanes 0-15 -> M=v, lanes 16-31 -> M=v+8; N = lane&15
    const int nOut = lane & 15;
    const int mOff = (lane < 16) ? 0 : 8;
    const int ocol = colBase + nOut;
    if (ocol < R) {
        const float bs = bias[ocol];
#pragma unroll
        for (int v = 0; v < 8; ++v) {
            const int bag = bagBase + mOff + v;
            if (bag < num_bags)
                out[(long)bag * R + ocol] = c[v] + bs;
        }
    }
}

extern "C" void kernel_launch(void* const* d_in, const int* in_sizes, int n_in,
                              void* d_out, int out_size, void* d_ws, size_t ws_size,
                              hipStream_t stream) {
    (void)n_in; (void)out_size; (void)d_ws; (void)ws_size;
    const float* repre  = (const float*)d_in[0];
    const float* relmat = (const float*)d_in[1];
    const float* bias   = (const float*)d_in[2];
    const int*   scope  = (const int*)d_in[3];
    const int*   labels = (const int*)d_in[4];
    float* out = (float*)d_out;

    const int num_bags = in_sizes[3] / 2;
    const int blocks   = (num_bags + BAGS_PER_BLOCK - 1) / BAGS_PER_BLOCK;

    bag_attention_fused<<<blocks, THREADS, 0, stream>>>(
        repre, relmat, bias, scope, labels, out, num_bags);
}